// DifferentiableNERF_83330955477580
// MI455X (gfx1250) — compile-verified
//
#include <hip/hip_runtime.h>
#include <hip/hip_bf16.h>
#include <math.h>

// DifferentiableNERF on gfx1250 (MI455X).
// Serial scan over L=512 residues; parallel over B=4096 chains.
// Latency-bound on the dependent place_dihedral chain -> run TWO independent
// chains per lane (in-lane ILP=2) so chain B fills chain A's VALU latency
// slots regardless of wave co-residency. LDS-staged output written back via
// CDNA5 GLOBAL_STORE_ASYNC_FROM_LDS_B32 (ASYNCcnt path, no VGPR transit).

#if __has_builtin(__builtin_amdgcn_global_store_async_from_lds_b32)
#define NERF_ASYNC_WB 1
#else
#define NERF_ASYNC_WB 0
#endif

namespace {

constexpr int L        = 512;
constexpr int STEPS    = L - 1;          // 511 scan steps
constexpr int ROWS     = 3 * L;          // 1536 output rows per chain
constexpr int CH       = 8;              // steps per chunk
constexpr int TPB      = 64;             // 2 wave32s per block
constexpr int CPL      = 2;              // chains per lane (ILP)
constexpr int CPB      = TPB * CPL;      // 128 chains per block
constexpr int OBSTRIDE = CH * 9 + 1;     // 73 floats; odd mod 64 banks -> conflict free

typedef __attribute__((address_space(1))) int as1_int;
typedef __attribute__((address_space(3))) int as3_int;

// Generic (flat) -> global AS1 pointer: flat VA of global memory is the
// global address itself.
__device__ __forceinline__ as1_int* to_global(void* p) {
  return reinterpret_cast<as1_int*>(reinterpret_cast<uintptr_t>(p));
}
// Generic (flat) -> LDS AS3 pointer: per ISA 10.2, the LDS address is the low 32 bits of the flat address,
// so truncating the flat address yields the LDS byte offset.
__device__ __forceinline__ as3_int* to_lds(void* p) {
  return reinterpret_cast<as3_int*>(
      (unsigned int)reinterpret_cast<uintptr_t>(p));
}

struct V3 { float x, y, z; };

__device__ __forceinline__ V3 vsub(V3 a, V3 b) { return {a.x - b.x, a.y - b.y, a.z - b.z}; }
__device__ __forceinline__ V3 vscale(V3 a, float s) { return {a.x * s, a.y * s, a.z * s}; }
__device__ __forceinline__ float vdot(V3 a, V3 b) { return a.x * b.x + a.y * b.y + a.z * b.z; }
__device__ __forceinline__ V3 vcross(V3 a, V3 b) {
  return {a.y * b.z - a.z * b.y, a.z * b.x - a.x * b.z, a.x * b.y - a.y * b.x};
}
// Matches reference _unit: x / (||x|| + 1e-8)
__device__ __forceinline__ V3 vunit(V3 a) {
  float n = sqrtf(vdot(a, a));
  float inv = 1.0f / (n + 1e-8f);
  return vscale(a, inv);
}

// Matches reference _place_dihedral exactly (including degenerate fallbacks).
__device__ __forceinline__ V3 place_dihedral(V3 a, V3 b, V3 c,
                                             float bond_angle, float bond_length,
                                             float torsion) {
  V3 ab = vsub(b, a);
  V3 bc = vunit(vsub(c, b));
  V3 cr = vcross(ab, bc);
  float crn = sqrtf(vdot(cr, cr));
  V3 n_main = vscale(cr, 1.0f / (crn + 1e-8f));

  // Fallback normal when ab x bc is degenerate.
  // ct = cross(bc, [1,0,0]) = (0, bc.z, -bc.y)
  float ctn = sqrtf(bc.z * bc.z + bc.y * bc.y);
  V3 perp2 = (ctn < 1e-6f) ? V3{0.0f, 1.0f, 0.0f} : V3{1.0f, 0.0f, 0.0f};
  V3 n_fb = vunit(vcross(bc, perp2));

  V3 n;
  n.x = (crn < 1e-6f) ? n_fb.x : n_main.x;
  n.y = (crn < 1e-6f) ? n_fb.y : n_main.y;
  n.z = (crn < 1e-6f) ? n_fb.z : n_main.z;

  V3 nbc = vcross(n, bc);

  float sa, ca, st, ct;
  __sincosf(bond_angle, &sa, &ca);
  __sincosf(torsion, &st, &ct);
  float d0 = -bond_length * ca;
  float d1 = bond_length * ct * sa;
  float d2 = bond_length * st * sa;

  // m = [bc | nbc | n] (columns), d = m @ d_local, result = d + c
  V3 d;
  d.x = bc.x * d0 + nbc.x * d1 + n.x * d2 + c.x;
  d.y = bc.y * d0 + nbc.y * d1 + n.y * d2 + c.y;
  d.z = bc.z * d0 + nbc.z * d1 + n.z * d2 + c.z;
  return d;
}

// Per-chain recurrence state + input bases.
struct Chain {
  V3 A, Bv, Cv;
  size_t srow;   // scalar stream base (phi/psi/omega)
  size_t vrow;   // vec3 stream base (lengths/angles)
  bool   act;
};

__device__ __forceinline__ void chain_init(Chain& c, int b, int B,
                                           float* out) {
  const int bq = (b < B) ? b : (B - 1);
  c.act  = (b < B);
  c.srow = (size_t)bq * L;
  c.vrow = (size_t)bq * L * 3;
  c.A  = {17.047f, 14.099f, 3.625f};   // N
  c.Bv = {16.967f, 12.784f, 4.338f};   // CA
  c.Cv = {15.685f, 12.755f, 5.133f};   // C
  if (c.act) {
    float* ob = out + (size_t)bq * (ROWS * 3);
    ob[0] = c.A.x;  ob[1] = c.A.y;  ob[2] = c.A.z;
    ob[3] = c.Bv.x; ob[4] = c.Bv.y; ob[5] = c.Bv.z;
    ob[6] = c.Cv.x; ob[7] = c.Cv.y; ob[8] = c.Cv.z;
  }
}

__device__ __forceinline__ void chain_step(Chain& ch, int i,
                                           const float* __restrict__ phi,
                                           const float* __restrict__ psi,
                                           const float* __restrict__ omega,
                                           const float* __restrict__ blen,
                                           const float* __restrict__ bang,
                                           float* __restrict__ r) {
  const float psi_i   = psi[ch.srow + i];
  const float omega_i = omega[ch.srow + i];
  const float phi_i   = phi[ch.srow + i + 1];

  const size_t vo = ch.vrow + (size_t)i * 3;
  const float len_nca = blen[vo + 0];
  const float len_cac = blen[vo + 1];
  const float len_cn  = blen[vo + 2];
  const float ang_ncac = bang[vo + 0];
  const float ang_cacn = bang[vo + 1];
  const float ang_cnca = bang[vo + 2];

  V3 n  = place_dihedral(ch.A,  ch.Bv, ch.Cv, ang_cacn, len_cn,  psi_i);
  V3 ca = place_dihedral(ch.Bv, ch.Cv, n,     ang_cnca, len_nca, omega_i);
  V3 cc = place_dihedral(ch.Cv, n,     ca,    ang_ncac, len_cac, phi_i);
  ch.A = n; ch.Bv = ca; ch.Cv = cc;

  r[0] = n.x;  r[1] = n.y;  r[2] = n.z;
  r[3] = ca.x; r[4] = ca.y; r[5] = ca.z;
  r[6] = cc.x; r[7] = cc.y; r[8] = cc.z;
}

__global__ __launch_bounds__(TPB)
void nerf_build_kernel(const float* __restrict__ phi,
                       const float* __restrict__ psi,
                       const float* __restrict__ omega,
                       const float* __restrict__ blen,   // [B, L, 3]
                       const float* __restrict__ bang,   // [B, L, 3]
                       float* __restrict__ out,          // [B, 1536, 3]
                       int B) {
  __shared__ float obuf[CPB * OBSTRIDE];  // 128 chains x 73 floats = 37,376 B

  const int t  = threadIdx.x;
  const int b0 = blockIdx.x * CPB;

  // Two independent chains per lane: rows t and t+TPB of obuf.
  Chain c0, c1;
  chain_init(c0, b0 + t,       B, out);
  chain_init(c1, b0 + TPB + t, B, out);

  float* const row0 = &obuf[t * OBSTRIDE];
  float* const row1 = &obuf[(TPB + t) * OBSTRIDE];

  for (int i0 = 0; i0 < STEPS; i0 += CH) {
    const int steps = (STEPS - i0 < CH) ? (STEPS - i0) : CH;

    // Prefetch next chunk's input lines into cache (global_prefetch_b8).
    const int inext = i0 + CH;
    if (inext < STEPS) {
      __builtin_prefetch(&psi[c0.srow + inext], 0, 0);
      __builtin_prefetch(&omega[c0.srow + inext], 0, 0);
      __builtin_prefetch(&phi[c0.srow + inext + 1], 0, 0);
      __builtin_prefetch(&blen[c0.vrow + (size_t)inext * 3], 0, 0);
      __builtin_prefetch(&bang[c0.vrow + (size_t)inext * 3], 0, 0);
      __builtin_prefetch(&psi[c1.srow + inext], 0, 0);
      __builtin_prefetch(&omega[c1.srow + inext], 0, 0);
      __builtin_prefetch(&phi[c1.srow + inext + 1], 0, 0);
      __builtin_prefetch(&blen[c1.vrow + (size_t)inext * 3], 0, 0);
      __builtin_prefetch(&bang[c1.vrow + (size_t)inext * 3], 0, 0);
    }

    for (int s = 0; s < steps; ++s) {
      const int i = i0 + s;
      // Independent recurrences: compiler interleaves them, hiding the
      // dependent-VALU latency of each behind the other.
      chain_step(c0, i, phi, psi, omega, blen, bang, &row0[s * 9]);
      chain_step(c1, i, phi, psi, omega, blen, bang, &row1[s * 9]);
    }

    __syncthreads();

    // Coalesced write-out: each chain's chunk is steps*9 contiguous floats.
    // Lanes stripe across a chain's region -> full 128B-line stores per wave.
    const int tot = steps * 9;
    const size_t gcol = (size_t)(3 + 3 * i0) * 3;  // float offset of chunk rows
    for (int bb = 0; bb < CPB; ++bb) {
      const int gb = b0 + bb;
      if (gb >= B) break;
      float* gdst = out + (size_t)gb * (ROWS * 3) + gcol;
      float* lsrc = &obuf[bb * OBSTRIDE];
      for (int k = t; k < tot; k += TPB) {
#if NERF_ASYNC_WB
        // CDNA5 ASYNCcnt path: LDS -> global with no VGPR transit.
        __builtin_amdgcn_global_store_async_from_lds_b32(
            to_global(gdst + k), to_lds(lsrc + k), 0, 0);
#else
        gdst[k] = lsrc[k];
#endif
      }
    }

#if NERF_ASYNC_WB
    // Drain this wave's async LDS reads before obuf is overwritten; after the
    // barrier every wave's ASYNCcnt has hit zero, so the buffer is safe.
#if __has_builtin(__builtin_amdgcn_s_wait_asynccnt)
    __builtin_amdgcn_s_wait_asynccnt(0);
#else
    asm volatile("s_wait_asynccnt 0x0" ::: "memory");
#endif
#endif

    __syncthreads();
  }
}

}  // namespace

extern "C" void kernel_launch(void* const* d_in, const int* in_sizes, int n_in,
                              void* d_out, int out_size, void* d_ws, size_t ws_size,
                              hipStream_t stream) {
  const float* phi   = (const float*)d_in[0];
  const float* psi   = (const float*)d_in[1];
  const float* omega = (const float*)d_in[2];
  const float* blen  = (const float*)d_in[3];
  const float* bang  = (const float*)d_in[4];
  float* out = (float*)d_out;

  const int B = in_sizes[0] / L;                  // phi is [B, L] flat
  const int grid = (B + CPB - 1) / CPB;

  hipLaunchKernelGGL(nerf_build_kernel, dim3(grid), dim3(TPB), 0, stream,
                     phi, psi, omega, blen, bang, out, B);
}